// SelfContraction_35777077575991
// MI455X (gfx1250) — compile-verified
//
#include <hip/hip_runtime.h>

typedef __attribute__((ext_vector_type(2))) float v2f;
typedef __attribute__((ext_vector_type(4))) float v4f;
typedef __attribute__((ext_vector_type(8))) float v8f;

#define NNODE 2048
#define CCH   128
#define R3    0.5773502691896258f   /* 1/sqrt(3) */
#define I3    0.3333333333333333f   /* 1/3       */

// symmetric-traceless projector (== reshape@D2 in the reference)
__device__ __forceinline__ void proj3(const float* A, float* P) {
    float tr = (A[0] + A[4] + A[8]) * I3;
    float s01 = 0.5f * (A[1] + A[3]);
    float s02 = 0.5f * (A[2] + A[6]);
    float s12 = 0.5f * (A[5] + A[7]);
    P[0] = A[0] - tr; P[4] = A[4] - tr; P[8] = A[8] - tr;
    P[1] = s01; P[3] = s01;
    P[2] = s02; P[6] = s02;
    P[5] = s12; P[7] = s12;
}

// ---------------------------------------------------------------------------
// Stage 1: per-(n,c) equivariant contractions -> X feature matrices
//   X0: [n][q0*128+c]              q0 in [0,15)  (K0 = 1920)
//   X1: [(n*3+t)][q1*128+c]        q1 in [0,20)  (K1 = 2560)
//   X2: [(n*9+t)][q2*128+c]        q2 in [0,20)  (K2 = 2560)
// ---------------------------------------------------------------------------
__global__ void sc_features_kernel(const float* __restrict__ f0,
                                   const float* __restrict__ f1,
                                   const float* __restrict__ f2,
                                   float* __restrict__ X0,
                                   float* __restrict__ X1,
                                   float* __restrict__ X2) {
    const int n  = blockIdx.x;
    const int c  = threadIdx.x;
    const int nc = n * CCH + c;

    float s = f0[nc];
    float v[3], M[9];
#pragma unroll
    for (int t = 0; t < 3; ++t) v[t] = f1[nc * 3 + t];
#pragma unroll
    for (int t = 0; t < 9; ++t) M[t] = f2[nc * 9 + t];

    auto wS = [&](int q, float x) { X0[(n * 15 + q) * CCH + c] = x; };
    auto wV = [&](int q, const float* x) {
#pragma unroll
        for (int t = 0; t < 3; ++t) X1[((n * 3 + t) * 20 + q) * CCH + c] = x[t];
    };
    auto wM = [&](int q, const float* x) {
#pragma unroll
        for (int t = 0; t < 9; ++t) X2[((n * 9 + t) * 20 + q) * CCH + c] = x[t];
    };

    // ---- v = 0 features (the raw inputs) ----
    wS(0, s); wV(0, v); wM(0, M);

    // ---- TMP[1] : first contraction round ----
    float S1[3];
    S1[0] = s * s;
    S1[1] = (v[0]*v[0] + v[1]*v[1] + v[2]*v[2]) * R3;
    {   float fr = 0.f;
#pragma unroll
        for (int t = 0; t < 9; ++t) fr += M[t] * M[t];
        S1[2] = fr * I3;
    }

    float V1[4][3];
#pragma unroll
    for (int t = 0; t < 3; ++t) { V1[0][t] = s * v[t]; V1[1][t] = s * v[t]; }
#pragma unroll
    for (int e = 0; e < 3; ++e)   // v^T M / sqrt3
        V1[2][e] = (v[0]*M[0*3+e] + v[1]*M[1*3+e] + v[2]*M[2*3+e]) * R3;
#pragma unroll
    for (int d = 0; d < 3; ++d)   // M v / sqrt3
        V1[3][d] = (M[d*3+0]*v[0] + M[d*3+1]*v[1] + M[d*3+2]*v[2]) * R3;

    float A1[4][9], tmp9[9];
#pragma unroll
    for (int t = 0; t < 9; ++t) tmp9[t] = s * M[t];
    proj3(tmp9, A1[0]);
#pragma unroll
    for (int t = 0; t < 9; ++t) A1[2][t] = A1[0][t];          // (2,0,2) path == (0,2,2)
#pragma unroll
    for (int i = 0; i < 3; ++i)
#pragma unroll
        for (int j = 0; j < 3; ++j) tmp9[i*3+j] = v[i] * v[j];
    proj3(tmp9, A1[1]);
#pragma unroll
    for (int d = 0; d < 3; ++d)                               // (M @ M) / sqrt3
#pragma unroll
        for (int f = 0; f < 3; ++f)
            tmp9[d*3+f] = (M[d*3+0]*M[0*3+f] + M[d*3+1]*M[1*3+f] + M[d*3+2]*M[2*3+f]) * R3;
    proj3(tmp9, A1[3]);

#pragma unroll
    for (int q = 0; q < 3; ++q) wS(1 + q, S1[q]);
#pragma unroll
    for (int p = 0; p < 4; ++p) wV(1 + p, V1[p]);
#pragma unroll
    for (int p = 0; p < 4; ++p) wM(1 + p, A1[p]);

    // ---- TMP[2] : second contraction round ----
    // scalars (11): (0,0,0)x3, (1,1,0)x4, (2,2,0)x4
#pragma unroll
    for (int q = 0; q < 3; ++q) wS(4 + q, S1[q] * s);
#pragma unroll
    for (int p = 0; p < 4; ++p)
        wS(7 + p, (V1[p][0]*v[0] + V1[p][1]*v[1] + V1[p][2]*v[2]) * R3);
#pragma unroll
    for (int p = 0; p < 4; ++p) {
        float fr = 0.f;
#pragma unroll
        for (int t = 0; t < 9; ++t) fr += A1[p][t] * M[t];
        wS(11 + p, fr * I3);
    }

    // vectors (15): (0,1,1)x3, (1,0,1)x4, (1,2,1)x4, (2,1,1)x4
    float vt[3];
#pragma unroll
    for (int q = 0; q < 3; ++q) {
#pragma unroll
        for (int t = 0; t < 3; ++t) vt[t] = S1[q] * v[t];
        wV(5 + q, vt);
    }
#pragma unroll
    for (int p = 0; p < 4; ++p) {
#pragma unroll
        for (int t = 0; t < 3; ++t) vt[t] = V1[p][t] * s;
        wV(8 + p, vt);
    }
#pragma unroll
    for (int p = 0; p < 4; ++p) {            // V1^T M / sqrt3
#pragma unroll
        for (int e = 0; e < 3; ++e)
            vt[e] = (V1[p][0]*M[0*3+e] + V1[p][1]*M[1*3+e] + V1[p][2]*M[2*3+e]) * R3;
        wV(12 + p, vt);
    }
#pragma unroll
    for (int p = 0; p < 4; ++p) {            // A1 v / sqrt3
#pragma unroll
        for (int d = 0; d < 3; ++d)
            vt[d] = (A1[p][d*3+0]*v[0] + A1[p][d*3+1]*v[1] + A1[p][d*3+2]*v[2]) * R3;
        wV(16 + p, vt);
    }

    // matrices (15): (0,2,2)x3, (1,1,2)x4, (2,0,2)x4, (2,2,2)x4
    float pm[9];
#pragma unroll
    for (int q = 0; q < 3; ++q) {
#pragma unroll
        for (int t = 0; t < 9; ++t) tmp9[t] = S1[q] * M[t];
        proj3(tmp9, pm);
        wM(5 + q, pm);
    }
#pragma unroll
    for (int p = 0; p < 4; ++p) {            // proj(V1 (x) v)
#pragma unroll
        for (int i = 0; i < 3; ++i)
#pragma unroll
            for (int j = 0; j < 3; ++j) tmp9[i*3+j] = V1[p][i] * v[j];
        proj3(tmp9, pm);
        wM(8 + p, pm);
    }
#pragma unroll
    for (int p = 0; p < 4; ++p) {            // proj(A1 * s) == A1 * s (idempotent)
#pragma unroll
        for (int t = 0; t < 9; ++t) pm[t] = A1[p][t] * s;
        wM(12 + p, pm);
    }
#pragma unroll
    for (int p = 0; p < 4; ++p) {            // proj(A1 @ M) / sqrt3
#pragma unroll
        for (int d = 0; d < 3; ++d)
#pragma unroll
            for (int f = 0; f < 3; ++f)
                tmp9[d*3+f] = (A1[p][d*3+0]*M[0*3+f] + A1[p][d*3+1]*M[1*3+f] +
                               A1[p][d*3+2]*M[2*3+f]) * R3;
        proj3(tmp9, pm);
        wM(16 + p, pm);
    }
}

// ---------------------------------------------------------------------------
// Stage 2: fold the two linears and TRANSPOSE for the WMMA B path:
//   WeffT_r[o][k] = sum_co Wcat_r[k][co] * Wo_r[co][o],  k = q*128+ci
// One block per k row (7040 rows), 128 threads = output channel o.
// ---------------------------------------------------------------------------
__global__ void sc_weff_kernel(const float* __restrict__ W00, const float* __restrict__ W01,
                               const float* __restrict__ W02, const float* __restrict__ W10,
                               const float* __restrict__ W11, const float* __restrict__ W12,
                               const float* __restrict__ W20, const float* __restrict__ W21,
                               const float* __restrict__ W22, const float* __restrict__ Wo0,
                               const float* __restrict__ Wo1, const float* __restrict__ Wo2,
                               float* __restrict__ We0, float* __restrict__ We1,
                               float* __restrict__ We2) {
    int flat = blockIdx.x;            // 0 .. 55*128-1
    int o    = threadIdx.x;
    int r, rem;
    if (flat < 15 * 128)      { r = 0; rem = flat; }
    else if (flat < 35 * 128) { r = 1; rem = flat - 15 * 128; }
    else                      { r = 2; rem = flat - 35 * 128; }
    int q  = rem >> 7;
    int ci = rem & 127;

    const float* Wsrc; const float* Wo; float* dst; int p, K;
    if (r == 0) {
        Wo = Wo0; dst = We0; K = 15 * CCH;
        if (q == 0)      { Wsrc = W00; p = 0; }
        else if (q < 4)  { Wsrc = W10; p = q - 1; }
        else             { Wsrc = W20; p = q - 4; }
    } else if (r == 1) {
        Wo = Wo1; dst = We1; K = 20 * CCH;
        if (q == 0)      { Wsrc = W01; p = 0; }
        else if (q < 5)  { Wsrc = W11; p = q - 1; }
        else             { Wsrc = W21; p = q - 5; }
    } else {
        Wo = Wo2; dst = We2; K = 20 * CCH;
        if (q == 0)      { Wsrc = W02; p = 0; }
        else if (q < 5)  { Wsrc = W12; p = q - 1; }
        else             { Wsrc = W22; p = q - 5; }
    }

    const float* wrow = Wsrc + (p * CCH + ci) * CCH;  // uniform per block -> scalar loads
    float acc = 0.f;
#pragma unroll 8
    for (int co = 0; co < CCH; ++co)
        acc = fmaf(wrow[co], Wo[co * CCH + o], acc);
    dst[o * K + q * CCH + ci] = acc;                  // transposed store [o][k]
}

// ---------------------------------------------------------------------------
// Stage 3: fin_r = X_r @ Weff_r + sc_r  via V_WMMA_F32_16X16X4_F32
// Block = 256 threads = 8 waves; block owns a 32-row M tile x 128 cols.
// Wave w owns output cols [16w, 16w+16) and TWO 16x16 accumulators (B reuse).
// LDS row pitch 132: bank-conflict-free halves + 16B-aligned rows.
// ---------------------------------------------------------------------------
template <int DIMS>
__global__ void sc_gemm_kernel(const float* __restrict__ X,
                               const float* __restrict__ WeffT,  // [128][K]
                               const float* __restrict__ sc,
                               float* __restrict__ out,
                               int K, int out_base) {
    __shared__ float Alds[32][132];

    const int m_base = blockIdx.x * 32;
    const int tid    = threadIdx.x;
    const int lane   = tid & 31;
    const int w      = tid >> 5;

    const int col16 = lane & 15;             // N index (and A row index)
    const int koff  = (lane >> 4) * 2;       // ISA 16x4 layout: lanes>=16 hold K=2,3
    const int o     = w * 16 + col16;

    v8f acc0 = {}, acc1 = {};
    const float* bbase = WeffT + o * K + koff;

    for (int kc = 0; kc < K; kc += 128) {
        // cooperative A fill: 32x128 chunk as float4 (b128 in, b128 to LDS)
#pragma unroll
        for (int i = 0; i < 4; ++i) {
            int e   = i * 1024 + tid * 4;
            int row = e >> 7;
            int col = e & 127;
            *(v4f*)&Alds[row][col] = *(const v4f*)&X[(m_base + row) * K + kc + col];
        }
        __syncthreads();

#pragma unroll 8
        for (int k0 = 0; k0 < 128; k0 += 4) {
            v2f a0 = *(const v2f*)&Alds[col16][k0 + koff];
            v2f a1 = *(const v2f*)&Alds[col16 + 16][k0 + koff];
            v2f b  = *(const v2f*)(bbase + kc + k0);
            acc0 = __builtin_amdgcn_wmma_f32_16x16x4_f32(
                false, a0, false, b, (short)0, acc0, false, false);
            acc1 = __builtin_amdgcn_wmma_f32_16x16x4_f32(
                false, a1, false, b, (short)0, acc1, false, false);
        }
        __syncthreads();
    }

    // epilogue: D layout -> vgpr i holds M = i + 8*(lane>=16), N = lane&15
    const int rowsel = (lane >> 4) << 3;
#pragma unroll
    for (int i = 0; i < 8; ++i) {
        int m = m_base + i + rowsel;
        int n = m / DIMS;
        int t = m - n * DIMS;
        out[n * 1664 + out_base + o * DIMS + t] =
            acc0[i] + sc[n * CCH * DIMS + o * DIMS + t];
        m += 16;
        n = m / DIMS;
        t = m - n * DIMS;
        out[n * 1664 + out_base + o * DIMS + t] =
            acc1[i] + sc[n * CCH * DIMS + o * DIMS + t];
    }
}

// ---------------------------------------------------------------------------
extern "C" void kernel_launch(void* const* d_in, const int* in_sizes, int n_in,
                              void* d_out, int out_size, void* d_ws, size_t ws_size,
                              hipStream_t stream) {
    const float* f0  = (const float*)d_in[0];
    const float* f1  = (const float*)d_in[1];
    const float* f2  = (const float*)d_in[2];
    /* d_in[3] = node_attrs : unused by the reference forward pass */
    const float* sc0 = (const float*)d_in[4];
    const float* sc1 = (const float*)d_in[5];
    const float* sc2 = (const float*)d_in[6];
    const float* W00 = (const float*)d_in[7];
    const float* W01 = (const float*)d_in[8];
    const float* W02 = (const float*)d_in[9];
    const float* W10 = (const float*)d_in[10];
    const float* W11 = (const float*)d_in[11];
    const float* W12 = (const float*)d_in[12];
    const float* W20 = (const float*)d_in[13];
    const float* W21 = (const float*)d_in[14];
    const float* W22 = (const float*)d_in[15];
    const float* Wo0 = (const float*)d_in[16];
    const float* Wo1 = (const float*)d_in[17];
    const float* Wo2 = (const float*)d_in[18];
    float* out = (float*)d_out;

    // workspace carve-up (floats)
    float* ws  = (float*)d_ws;
    float* X0  = ws;                                    // 2048*1920
    float* X1  = X0 + (size_t)NNODE * 1920;             // 2048*3*2560
    float* X2  = X1 + (size_t)NNODE * 3 * 2560;         // 2048*9*2560
    float* We0 = X2 + (size_t)NNODE * 9 * 2560;         // 128*1920 (transposed)
    float* We1 = We0 + (size_t)CCH * 1920;              // 128*2560
    float* We2 = We1 + (size_t)CCH * 2560;              // 128*2560

    sc_features_kernel<<<NNODE, CCH, 0, stream>>>(f0, f1, f2, X0, X1, X2);
    sc_weff_kernel<<<55 * 128, CCH, 0, stream>>>(W00, W01, W02, W10, W11, W12,
                                                 W20, W21, W22, Wo0, Wo1, Wo2,
                                                 We0, We1, We2);

    // fin_r = X_r @ Weff_r + sc_r  -> concatenated output rows of width 1664
    sc_gemm_kernel<1><<< NNODE * 1 / 32, 256, 0, stream>>>(X0, We0, sc0, out, 1920, 0);
    sc_gemm_kernel<3><<< NNODE * 3 / 32, 256, 0, stream>>>(X1, We1, sc1, out, 2560, 128);
    sc_gemm_kernel<9><<< NNODE * 9 / 32, 256, 0, stream>>>(X2, We2, sc2, out, 2560, 512);
}